// ExpLambsMessageAggregator_56238301774531
// MI455X (gfx1250) — compile-verified
//
#include <hip/hip_runtime.h>
#include <math.h>

typedef __attribute__((ext_vector_type(2))) float v2f;
typedef __attribute__((ext_vector_type(8))) float v8f;

#define N_NODES 65536
#define DIM 128
#define KDIM 4

// One wave (32 lanes) per node. Segment GEMM agg[n] = W^T (E_seg x 4) * M (E_seg x 128)
// via v_wmma_f32_16x16x4_f32: 4 edges per WMMA, 8 d-chunks of 16 accumulated in 8 tiles.
__global__ __launch_bounds__(256)
void ExpLambsMessageAggregator_kernel(const float* __restrict__ messages,
                                      const float* __restrict__ timestamps,
                                      const int*   __restrict__ seg_ids,
                                      const float* __restrict__ lambs,
                                      float* __restrict__ out_agg,
                                      float* __restrict__ out_tlast,
                                      int E, int n_nodes)
{
    const int lane = threadIdx.x & 31;
    const int wave = __builtin_amdgcn_readfirstlane((int)(threadIdx.x >> 5));
    const int node = blockIdx.x * 8 + wave;
    if (node >= n_nodes) return;               // wave-uniform branch

    // Uniform binary searches: [start, end) = edge range of this node's segment.
    int lo = 0, hi = E;
    while (lo < hi) { int mid = (lo + hi) >> 1; if (seg_ids[mid] < node)     lo = mid + 1; else hi = mid; }
    const int start = lo;
    hi = E;
    while (lo < hi) { int mid = (lo + hi) >> 1; if (seg_ids[mid] < node + 1) lo = mid + 1; else hi = mid; }
    const int end = lo;

    const int h = lane >> 4;                   // lane half: edge pair selector
    const int c = lane & 15;                   // A: output-k row (c<4 useful); B: d column
    const float t_last   = (end > start) ? timestamps[end - 1] : -INFINITY;
    const float inv_lamb = 1.0f / lambs[c & 3];

    v8f acc[8];
    #pragma unroll
    for (int i = 0; i < 8; ++i) acc[i] = (v8f){0.f,0.f,0.f,0.f,0.f,0.f,0.f,0.f};

    for (int g = start; g < end; g += 4) {     // 4 edges per WMMA group (uniform loop)
        const int e0 = g + 2 * h;              // this half-wave's two edges
        const int e1 = e0 + 1;
        const int e0c = min(e0, end - 1);      // clamped addresses (weight is 0 when OOB)
        const int e1c = min(e1, end - 1);

        const float t0 = timestamps[e0c];
        const float t1 = timestamps[e1c];
        v2f a;                                 // A[m=c][K=2h+v] = w(edge, k=c), rows c>=4 -> 0
        a.x = (c < 4 && e0 < end) ? __expf((t0 - t_last) * inv_lamb) : 0.0f;
        a.y = (c < 4 && e1 < end) ? __expf((t1 - t_last) * inv_lamb) : 0.0f;

        const float* __restrict__ r0 = messages + (size_t)e0c * DIM;
        const float* __restrict__ r1 = messages + (size_t)e1c * DIM;

        #pragma unroll
        for (int ch = 0; ch < 8; ++ch) {       // 8 d-chunks of 16
            v2f b;                             // B[K=2h+v][n=c] = messages[e][ch*16 + c]
            b.x = __builtin_nontemporal_load(&r0[ch * 16 + c]);
            b.y = __builtin_nontemporal_load(&r1[ch * 16 + c]);
            // 8 args: (neg_a, A, neg_b, B, c_mod, C, reuse_a, reuse_b)
            acc[ch] = __builtin_amdgcn_wmma_f32_16x16x4_f32(
                false, a, false, b, (short)0, acc[ch], false, false);
        }
    }

    // D tile: VGPR r, lanes 0-15 hold out[M=r][N=lane] -> out[k=r][d=ch*16+lane], r<4 useful.
    if (h == 0) {
        float* __restrict__ base = out_agg + (size_t)node * (KDIM * DIM);
        #pragma unroll
        for (int ch = 0; ch < 8; ++ch) {
            __builtin_nontemporal_store(acc[ch][0], &base[0 * DIM + ch * 16 + c]);
            __builtin_nontemporal_store(acc[ch][1], &base[1 * DIM + ch * 16 + c]);
            __builtin_nontemporal_store(acc[ch][2], &base[2 * DIM + ch * 16 + c]);
            __builtin_nontemporal_store(acc[ch][3], &base[3 * DIM + ch * 16 + c]);
        }
    }
    if (lane == 0) out_tlast[node] = t_last;
}

extern "C" void kernel_launch(void* const* d_in, const int* in_sizes, int n_in,
                              void* d_out, int out_size, void* d_ws, size_t ws_size,
                              hipStream_t stream) {
    const float* messages   = (const float*)d_in[0];
    const float* timestamps = (const float*)d_in[1];
    const int*   seg_ids    = (const int*)  d_in[2];
    const float* lambs      = (const float*)d_in[3];
    const int E = in_sizes[1];                       // timestamps element count

    float* out_agg   = (float*)d_out;                // [N, K, D]
    float* out_tlast = out_agg + (size_t)N_NODES * (KDIM * DIM);  // [N]

    dim3 grid(N_NODES / 8), block(256);              // 8 waves/block, 1 node/wave
    hipLaunchKernelGGL(ExpLambsMessageAggregator_kernel, grid, block, 0, stream,
                       messages, timestamps, seg_ids, lambs,
                       out_agg, out_tlast, E, N_NODES);
}